// GAT_52776558133401
// MI455X (gfx1250) — compile-verified
//
#include <hip/hip_runtime.h>
#include <math.h>

typedef __attribute__((ext_vector_type(2))) float        v2f;
typedef __attribute__((ext_vector_type(8))) float        v8f;
typedef __attribute__((ext_vector_type(4))) unsigned int v4u;
typedef __attribute__((ext_vector_type(8))) int          v8i;
typedef __attribute__((ext_vector_type(4))) int          v4i;

#define NEG_SLOPE 0.2f

// ---------------------------------------------------------------------------
// TDM: issue TENSOR_LOAD_TO_LDS for a 2D f32 tile (CDNA5 ISA).
//   group0: count=1 | lds_addr | global_addr[56:0] | type=2
//   group1: data_size=4B, optional LDS row padding, dims/tiles/stride0
//   remaining groups: zero (<=2D tensor)
// This toolchain's builtin is the 6-arg clang-23 form:
//   (uint32x4, int32x8, int32x4, int32x4, int32x8, i32 cpol)
// ---------------------------------------------------------------------------
__device__ __forceinline__ unsigned lds_off(const void* p)
{
    // LDS aperture: LDS address = flat_addr[31:0]
    return (unsigned)(uintptr_t)p;
}

__device__ __forceinline__ void tdm_load_2d_f32(
    const void* gaddr, unsigned lds_byte_addr,
    unsigned dim0, unsigned dim1,            // tensor extent (elements): contig, rows
    unsigned tile0, unsigned tile1,          // tile extent (elements)
    unsigned long long stride0,              // row stride (elements)
    unsigned pad_interval_code, unsigned pad_amount_code, bool pad_en)
{
    unsigned long long ga = (unsigned long long)(uintptr_t)gaddr;
    v4u g0;
    g0[0] = 1u;                                              // count=1 (user D#)
    g0[1] = lds_byte_addr;                                   // bits 63:32
    g0[2] = (unsigned)(ga & 0xFFFFFFFFu);                    // global_addr[31:0]
    g0[3] = (unsigned)((ga >> 32) & 0x01FFFFFFu)             // global_addr[56:32]
          | (2u << 30);                                      // type=2 ("image")
    unsigned d0 = (2u << 16);                                // data_size: 4 bytes
    if (pad_en)
        d0 |= (1u << 20) | (pad_interval_code << 22) | (pad_amount_code << 25);
    v8i g1;
    g1[0] = (int)d0;                                         // wg_mask=0 (no cluster)
    g1[1] = (int)((dim0 & 0xFFFFu) << 16);                   // tensor_dim0[15:0] @63:48
    g1[2] = (int)(((dim0 >> 16) & 0xFFFFu) | ((dim1 & 0xFFFFu) << 16));
    g1[3] = (int)(((dim1 >> 16) & 0xFFFFu) | ((tile0 & 0xFFFFu) << 16));
    g1[4] = (int)(tile1 & 0xFFFFu);                          // tile_dim1; tile_dim2=0
    g1[5] = (int)(unsigned)(stride0 & 0xFFFFFFFFull);        // tensor_dim0_stride[31:0]
    g1[6] = (int)(unsigned)((stride0 >> 32) & 0xFFFFull);    // stride[47:32]; dim1_stride=0
    g1[7] = 0;
    v4i z4 = {0, 0, 0, 0};
    v8i z8 = {0, 0, 0, 0, 0, 0, 0, 0};
    __builtin_amdgcn_tensor_load_to_lds(g0, g1, z4, z4, z8, 0);
}

// ---------------------------------------------------------------------------
// C[N,M] = A[N,K] @ B[K,M] in fp32 via V_WMMA_F32_16X16X4_F32, operands staged
// in LDS by the Tensor Data Mover. Block = 8 waves = 128-row stripe; B (whole
// weight matrix, K=128 rows) is loaded once per block. A stripe is loaded with
// a 4-DWORD pad per 128-DWORD row so the 16 lanes of an A-fragment read hit
// distinct LDS bank groups. K must be 128 (pad_interval code 6).
// ---------------------------------------------------------------------------
template <int TM>
__global__ __launch_bounds__(256) void gemm_f32_wmma_lds(
    const float* __restrict__ A, const float* __restrict__ B,
    float* __restrict__ C, int Nrows, int K, int M)
{
    extern __shared__ float smem[];
    const int KP   = K + 4;                    // padded A row stride (floats)
    float*  Alds   = smem;                     // 128 x KP
    float*  Blds   = smem + 128 * KP;          // K x M

    const int rowsLeft  = Nrows - blockIdx.x * 128;
    const int blockRows = rowsLeft < 128 ? rowsLeft : 128;   // multiple of 16

    if (threadIdx.x < 32) {                    // wave 0 issues the DMAs
        const float* Ag = A + (size_t)blockIdx.x * 128 * K;
        __builtin_prefetch(Ag, 0, 0);          // global_prefetch_b8
        tdm_load_2d_f32(Ag, lds_off(Alds), (unsigned)K, (unsigned)blockRows,
                        (unsigned)K, (unsigned)blockRows, (unsigned long long)K,
                        /*pad_interval: 128 dw*/ 6u, /*pad: 4 dw*/ 3u, true);
        tdm_load_2d_f32(B, lds_off(Blds), (unsigned)M, (unsigned)K,
                        (unsigned)M, (unsigned)K, (unsigned long long)M,
                        0u, 0u, false);
        __builtin_amdgcn_s_wait_tensorcnt(0);
    }
    __syncthreads();

    const int wv   = threadIdx.x >> 5;
    const int lane = threadIdx.x & 31;
    const int half = lane >> 4;
    const int l16  = lane & 15;
    const int row0 = wv * 16;                  // within the block stripe

    if (row0 < blockRows) {                    // wave-uniform: EXEC stays all-1s
        const float* Arow = Alds + (size_t)(row0 + l16) * KP;
        const int nG = M / (16 * TM);
        for (int g = 0; g < nG; ++g) {
            const int c0 = g * (16 * TM) + l16;
            v8f acc[TM] = {};
            for (int k = 0; k < K; k += 4) {
                const int ka = k + 2 * half;   // A frag: lane<16 -> K k,k+1 ; else k+2,k+3
                v2f a;
                a.x = Arow[ka];
                a.y = Arow[ka + 1];
#pragma unroll
                for (int t = 0; t < TM; ++t) { // B frag: rows ka, ka+1 striped over lanes
                    v2f b;
                    b.x = Blds[(size_t)ka       * M + c0 + t * 16];
                    b.y = Blds[(size_t)(ka + 1) * M + c0 + t * 16];
                    acc[t] = __builtin_amdgcn_wmma_f32_16x16x4_f32(
                        false, a, false, b, (short)0, acc[t], false, false);
                }
            }
            float* Crow = C + (size_t)(blockIdx.x * 128 + row0 + half * 8) * M;
#pragma unroll
            for (int t = 0; t < TM; ++t)
#pragma unroll
                for (int v = 0; v < 8; ++v)    // C/D: VGPR v = row v+8*half
                    Crow[(size_t)v * M + c0 + t * 16] = acc[t][v];
        }
    }
}

// --------------------------------------------------------------------------
// al_src[n,h] = sum_c h[n,h,c]*a_src[h,c] ; same for a_dst. One thread/(n,h).
// --------------------------------------------------------------------------
__global__ void attn_logits_kernel(const float* __restrict__ h,
                                   const float* __restrict__ a_src,
                                   const float* __restrict__ a_dst,
                                   float* __restrict__ als, float* __restrict__ ald,
                                   int Nn, int H, int C)
{
    int i = blockIdx.x * blockDim.x + threadIdx.x;
    if (i >= Nn * H) return;
    int n = i / H, hh = i % H;
    const float* hp  = h + (size_t)n * H * C + (size_t)hh * C;
    const float* asv = a_src + hh * C;
    const float* adv = a_dst + hh * C;
    float s1 = 0.f, s2 = 0.f;
    for (int c = 0; c < C; ++c) { float v = hp[c]; s1 += v * asv[c]; s2 += v * adv[c]; }
    als[i] = s1; ald[i] = s2;
}

__global__ void init_bufs_kernel(float* __restrict__ m, float* __restrict__ dn,
                                 float* __restrict__ agg, int nm, int nagg)
{
    int i = blockIdx.x * blockDim.x + threadIdx.x;
    if (i < nm)   { m[i] = -__builtin_inff(); dn[i] = 0.f; }
    if (i < nagg) { agg[i] = 0.f; }
}

__device__ __forceinline__ void edge_sd(const long long* __restrict__ ei, int e, int E,
                                        int& s, int& d)
{
    if (e < E) { s = (int)ei[e]; d = (int)ei[(size_t)E + e]; }
    else       { s = e - E;      d = e - E; }                 // self-loops
}

// monotonic-int-bits float atomic max (values are finite, init is -inf)
__device__ __forceinline__ void atomicMaxF(float* addr, float val)
{
    if (val >= 0.f) atomicMax((int*)addr, __float_as_int(val));
    else            atomicMin((unsigned int*)addr, __float_as_uint(val));
}

// e = leaky_relu(al_src[src]+al_dst[dst]) ; running segment max over dst
__global__ void edge_logits_max_kernel(const long long* __restrict__ ei,
                                       const float* __restrict__ als,
                                       const float* __restrict__ ald,
                                       float* __restrict__ eb, float* __restrict__ m,
                                       int E, int Etot, int H)
{
    long long i = (long long)blockIdx.x * blockDim.x + threadIdx.x;
    if (i >= (long long)Etot * H) return;
    int e = (int)(i / H), hh = (int)(i % H);
    int s, d; edge_sd(ei, e, E, s, d);
    float v = als[(size_t)s * H + hh] + ald[(size_t)d * H + hh];
    v = (v > 0.f) ? v : NEG_SLOPE * v;
    eb[i] = v;
    atomicMaxF(&m[(size_t)d * H + hh], v);
}

__global__ void edge_expsum_kernel(const long long* __restrict__ ei,
                                   float* __restrict__ eb,
                                   const float* __restrict__ m, float* __restrict__ dn,
                                   int E, int Etot, int H)
{
    long long i = (long long)blockIdx.x * blockDim.x + threadIdx.x;
    if (i >= (long long)Etot * H) return;
    int e = (int)(i / H), hh = (int)(i % H);
    int s, d; edge_sd(ei, e, E, s, d);
    float ex = __expf(eb[i] - m[(size_t)d * H + hh]);
    eb[i] = ex;
    atomicAdd(&dn[(size_t)d * H + hh], ex);
}

__global__ void edge_alpha_kernel(const long long* __restrict__ ei,
                                  float* __restrict__ eb, const float* __restrict__ dn,
                                  int E, int Etot, int H)
{
    long long i = (long long)blockIdx.x * blockDim.x + threadIdx.x;
    if (i >= (long long)Etot * H) return;
    int e = (int)(i / H), hh = (int)(i % H);
    int s, d; edge_sd(ei, e, E, s, d);
    eb[i] = eb[i] / dn[(size_t)d * H + hh];
}

// agg[dst, f..f+3] += alpha[e, f/C] * h[src, f..f+3]  (float4 gather per thread;
// a 4-aligned feature block never crosses a head boundary for C in {8,32})
__global__ void edge_aggregate_kernel(const long long* __restrict__ ei,
                                      const float* __restrict__ h,
                                      const float* __restrict__ alpha,
                                      float* __restrict__ agg,
                                      int E, int Etot, int H, int C)
{
    const int HC  = H * C;
    const int HC4 = HC >> 2;
    long long i = (long long)blockIdx.x * blockDim.x + threadIdx.x;
    if (i >= (long long)Etot * HC4) return;
    int e = (int)(i / HC4), f = (int)(i % HC4) << 2;
    int s, d; edge_sd(ei, e, E, s, d);
    float  al = alpha[(size_t)e * H + f / C];
    float4 hv = *(const float4*)(h + (size_t)s * HC + f);
    float* ap = agg + (size_t)d * HC + f;
    atomicAdd(ap + 0, al * hv.x);
    atomicAdd(ap + 1, al * hv.y);
    atomicAdd(ap + 2, al * hv.z);
    atomicAdd(ap + 3, al * hv.w);
}

__global__ void bias_relu_kernel(float* __restrict__ a, const float* __restrict__ bias,
                                 int Nn, int F)
{
    long long i = (long long)blockIdx.x * blockDim.x + threadIdx.x;
    if (i >= (long long)Nn * F) return;
    float v = a[i] + bias[(int)(i % F)];
    a[i] = fmaxf(v, 0.f);
}

// out[n,:] = log_softmax(in[n,:] + bias)   (F = 32; one thread per node)
__global__ void log_softmax_kernel(const float* __restrict__ in,
                                   const float* __restrict__ bias,
                                   float* __restrict__ out, int Nn, int F)
{
    int n = blockIdx.x * blockDim.x + threadIdx.x;
    if (n >= Nn) return;
    const float* r = in + (size_t)n * F;
    float mx = -__builtin_inff();
    for (int j = 0; j < F; ++j) mx = fmaxf(mx, r[j] + bias[j]);
    float s = 0.f;
    for (int j = 0; j < F; ++j) s += __expf(r[j] + bias[j] - mx);
    float ls = __logf(s);
    for (int j = 0; j < F; ++j) out[(size_t)n * F + j] = r[j] + bias[j] - mx - ls;
}

static inline unsigned cdiv_ll(long long a, long long b) { return (unsigned)((a + b - 1) / b); }

extern "C" void kernel_launch(void* const* d_in, const int* in_sizes, int n_in,
                              void* d_out, int out_size, void* d_ws, size_t ws_size,
                              hipStream_t stream)
{
    const float*     x   = (const float*)d_in[0];
    const long long* ei  = (const long long*)d_in[1];
    const float*     W1  = (const float*)d_in[2];
    const float*     as1 = (const float*)d_in[3];
    const float*     ad1 = (const float*)d_in[4];
    const float*     b1  = (const float*)d_in[5];
    const float*     W2  = (const float*)d_in[6];
    const float*     as2 = (const float*)d_in[7];
    const float*     ad2 = (const float*)d_in[8];
    const float*     b2  = (const float*)d_in[9];
    float* out = (float*)d_out;

    const int Fin = 128, H = 4, C1 = 32, HC1 = 128, C2 = 8, HC2 = 32;
    const int N    = in_sizes[0] / Fin;
    const int E    = in_sizes[1] / 2;
    const int Etot = E + N;

    float* ws   = (float*)d_ws;
    float* h1   = ws;  ws += (size_t)N * HC1;
    float* agg1 = ws;  ws += (size_t)N * HC1;
    float* als  = ws;  ws += (size_t)N * H;
    float* ald  = ws;  ws += (size_t)N * H;
    float* mb   = ws;  ws += (size_t)N * H;
    float* dn   = ws;  ws += (size_t)N * H;
    float* eb   = ws;  ws += (size_t)Etot * H;
    float* h2   = ws;  ws += (size_t)N * HC2;
    float* agg2 = ws;

    const dim3 blk(256);
    const unsigned gemmBlocks = cdiv_ll(N, 128);
    const size_t shmem1 = (size_t)(128 * (Fin + 4) + Fin * HC1) * sizeof(float); // 133 KB
    const size_t shmem2 = (size_t)(128 * (HC1 + 4) + HC1 * HC2) * sizeof(float); //  84 KB

    // ---------------- layer 1 ----------------
    gemm_f32_wmma_lds<4><<<gemmBlocks, blk, shmem1, stream>>>(x, W1, h1, N, Fin, HC1);
    attn_logits_kernel<<<cdiv_ll((long long)N * H, 256), blk, 0, stream>>>(
        h1, as1, ad1, als, ald, N, H, C1);
    init_bufs_kernel<<<cdiv_ll((long long)N * HC1, 256), blk, 0, stream>>>(
        mb, dn, agg1, N * H, N * HC1);
    edge_logits_max_kernel<<<cdiv_ll((long long)Etot * H, 256), blk, 0, stream>>>(
        ei, als, ald, eb, mb, E, Etot, H);
    edge_expsum_kernel<<<cdiv_ll((long long)Etot * H, 256), blk, 0, stream>>>(
        ei, eb, mb, dn, E, Etot, H);
    edge_alpha_kernel<<<cdiv_ll((long long)Etot * H, 256), blk, 0, stream>>>(
        ei, eb, dn, E, Etot, H);
    edge_aggregate_kernel<<<cdiv_ll((long long)Etot * (HC1 / 4), 256), blk, 0, stream>>>(
        ei, h1, eb, agg1, E, Etot, H, C1);
    bias_relu_kernel<<<cdiv_ll((long long)N * HC1, 256), blk, 0, stream>>>(
        agg1, b1, N, HC1);

    // ---------------- layer 2 ----------------
    gemm_f32_wmma_lds<2><<<gemmBlocks, blk, shmem2, stream>>>(agg1, W2, h2, N, HC1, HC2);
    attn_logits_kernel<<<cdiv_ll((long long)N * H, 256), blk, 0, stream>>>(
        h2, as2, ad2, als, ald, N, H, C2);
    init_bufs_kernel<<<cdiv_ll((long long)N * HC2, 256), blk, 0, stream>>>(
        mb, dn, agg2, N * H, N * HC2);
    edge_logits_max_kernel<<<cdiv_ll((long long)Etot * H, 256), blk, 0, stream>>>(
        ei, als, ald, eb, mb, E, Etot, H);
    edge_expsum_kernel<<<cdiv_ll((long long)Etot * H, 256), blk, 0, stream>>>(
        ei, eb, mb, dn, E, Etot, H);
    edge_alpha_kernel<<<cdiv_ll((long long)Etot * H, 256), blk, 0, stream>>>(
        ei, eb, dn, E, Etot, H);
    edge_aggregate_kernel<<<cdiv_ll((long long)Etot * (HC2 / 4), 256), blk, 0, stream>>>(
        ei, h2, eb, agg2, E, Etot, H, C2);
    log_softmax_kernel<<<cdiv_ll(N, 256), blk, 0, stream>>>(agg2, b2, out, N, HC2);
}